// EdgeGCN_K_Set2Set_13039520710682
// MI455X (gfx1250) — compile-verified
//
#include <hip/hip_runtime.h>
#include <hip/hip_bf16.h>
#include <math.h>

// Problem dims (fixed by the reference)
#define NN 32768
#define EE 524288
#define HH 128
#define BB 64
#define TT 12
#define LL 3
#define NODE_FD 64
#define EDGE_FD 16

typedef float v2f __attribute__((ext_vector_type(2)));
typedef float v8f __attribute__((ext_vector_type(8)));

__device__ __forceinline__ float sigf(float x) { return 1.0f / (1.0f + expf(-x)); }

// monotone float <-> uint mapping so unsigned atomicMax == float max
__device__ __forceinline__ unsigned fenc(float f) {
  unsigned u = __float_as_uint(f);
  return (u & 0x80000000u) ? ~u : (u | 0x80000000u);
}
__device__ __forceinline__ float fdec(unsigned k) {
  unsigned u = (k & 0x80000000u) ? (k & 0x7FFFFFFFu) : ~k;
  return __uint_as_float(u);
}

// ---------------------------------------------------------------------------
// Y[N x 128] = A[N x K] @ W[K x 128] (+ bias), f32 WMMA 16x16x4.
// Block = 8 waves. blockIdx.y selects a 16-col tile of W (staged in LDS,
// shared by all 8 waves); each wave owns a distinct 16-row tile of A.
// A-frag per ISA layout: lane&15 = M, (lane>>4)*2 = K offset within k-step.
// C/D layout: VGPR v -> row v (lanes 0-15) / row v+8 (lanes 16-31), col=lane&15.
// ---------------------------------------------------------------------------
template <int K>
__global__ __launch_bounds__(256) void gemm_xw_wmma(
    const float* __restrict__ A, const float* __restrict__ W,
    const float* __restrict__ bias, float* __restrict__ Y) {
  __shared__ float Bs[K * 16];
  const int tid = threadIdx.x;
  const int n0 = blockIdx.y << 4;
  for (int i = tid; i < K * 16; i += 256) {
    Bs[i] = W[(i >> 4) * HH + n0 + (i & 15)];  // Bs[k][n], coalesced per 16 lanes
  }
  __syncthreads();

  const int lane = tid & 31;
  const int nloc = lane & 15;        // M for A-frag, N for B/C-frags
  const int hi = lane >> 4;          // half-wave select
  const int m0 = (blockIdx.x * 8 + (tid >> 5)) << 4;

  const float* arow = A + (size_t)(m0 + nloc) * K;
  v8f acc = {};
#pragma unroll
  for (int k0 = 0; k0 < K; k0 += 4) {
    const int ka = k0 + (hi << 1);
    v2f a;
    a.x = arow[ka];
    a.y = arow[ka + 1];
    v2f b;
    b.x = Bs[ka * 16 + nloc];
    b.y = Bs[(ka + 1) * 16 + nloc];
    acc = __builtin_amdgcn_wmma_f32_16x16x4_f32(false, a, false, b,
                                                (short)0, acc, false, false);
  }
  const float bv = bias ? bias[n0 + nloc] : 0.0f;
#pragma unroll
  for (int v = 0; v < 8; ++v) {
    Y[(size_t)(m0 + v + (hi << 3)) * HH + n0 + nloc] = acc[v] + bv;
  }
}

// ---------------------------------------------------------------------------
// Per-edge: msg = Y[Esrc] * sigmoid(EF[e] @ Wee + bee); atomicAdd into Acc[Etgt].
// Wave = 1 edge, lane = 4 channels (float4). Gate recomputed from the 16 edge
// features + LDS-resident Wee — saves ~1 GB of ef traffic vs materializing.
// Gather/scatter target buffers are 16 MB each -> resident in 192 MB L2.
// ---------------------------------------------------------------------------
__global__ __launch_bounds__(256) void edge_msg(
    const float* __restrict__ Y, const float* __restrict__ EF,
    const int* __restrict__ Esrc, const int* __restrict__ Etgt,
    const float* __restrict__ Wee, const float* __restrict__ bee,
    float* __restrict__ Acc) {
  __shared__ float WeeS[EDGE_FD * HH];
  __shared__ float beeS[HH];
  for (int i = threadIdx.x; i < EDGE_FD * HH; i += 256) WeeS[i] = Wee[i];
  if (threadIdx.x < HH) beeS[threadIdx.x] = bee[threadIdx.x];
  __syncthreads();

  const int lane = threadIdx.x & 31;
  const int e = blockIdx.x * 8 + (threadIdx.x >> 5);   // E divides evenly
  const int src = Esrc[e];
  const int tgt = Etgt[e];
  const int c0 = lane * 4;

  const float4 yv = *(const float4*)(Y + (size_t)src * HH + c0);

  float efv[EDGE_FD];
  const float4* efp = (const float4*)(EF + (size_t)e * EDGE_FD);
#pragma unroll
  for (int q = 0; q < 4; ++q) {
    float4 f = efp[q];
    efv[4 * q + 0] = f.x; efv[4 * q + 1] = f.y;
    efv[4 * q + 2] = f.z; efv[4 * q + 3] = f.w;
  }

  float m[4] = {yv.x, yv.y, yv.z, yv.w};
#pragma unroll
  for (int j = 0; j < 4; ++j) {
    const int c = c0 + j;
    float s = beeS[c];
#pragma unroll
    for (int k = 0; k < EDGE_FD; ++k) s += efv[k] * WeeS[k * HH + c];
    m[j] *= sigf(s);
  }
  float* dst = Acc + (size_t)tgt * HH + c0;
  atomicAdd(dst + 0, m[0]);
  atomicAdd(dst + 1, m[1]);
  atomicAdd(dst + 2, m[2]);
  atomicAdd(dst + 3, m[3]);
}

// x = (relu?) (Acc + bg[l])
__global__ __launch_bounds__(256) void bias_act(
    const float* __restrict__ Acc, const float* __restrict__ bgl,
    float* __restrict__ X, int do_relu) {
  const int total = NN * HH;
  for (int i = blockIdx.x * blockDim.x + threadIdx.x; i < total;
       i += gridDim.x * blockDim.x) {
    float v = Acc[i] + bgl[i & (HH - 1)];
    X[i] = do_relu ? fmaxf(v, 0.0f) : v;
  }
}

// ---------------------------------------------------------------------------
// LSTM cell: thread = (b, j). Computes all 4 gates for its channel (K=256+128
// dots, ~25 MFLOP total — tiny), updates c in place, writes h to the ping-pong
// buffer (full-row h reads forbid in-place). Also resets the per-step
// softmax accumulators so no extra init kernel is needed.
// ---------------------------------------------------------------------------
__global__ __launch_bounds__(256) void lstm_step(
    const float* __restrict__ qstar, const float* __restrict__ h_old,
    float* __restrict__ h_new, float* __restrict__ cst,
    const float* __restrict__ Wih, const float* __restrict__ Whh,
    const float* __restrict__ bih, const float* __restrict__ bhh,
    unsigned* __restrict__ emax, float* __restrict__ ssum,
    float* __restrict__ rnum) {
  const int idx = blockIdx.x * 256 + threadIdx.x;  // 0..8191
  const int b = idx >> 7;
  const int j = idx & (HH - 1);
  if (idx < BB) { emax[idx] = 0u; ssum[idx] = 0.0f; }
  rnum[idx] = 0.0f;

  const float* q = qstar + (size_t)b * 2 * HH;
  const float* h = h_old + (size_t)b * HH;
  float g4[4];
#pragma unroll
  for (int gi = 0; gi < 4; ++gi) {
    const int jj = gi * HH + j;
    float s = bih[jj] + bhh[jj];
    const float* wi = Wih + (size_t)jj * (2 * HH);
    for (int k = 0; k < 2 * HH; ++k) s += q[k] * wi[k];
    const float* wh = Whh + (size_t)jj * HH;
    for (int k = 0; k < HH; ++k) s += h[k] * wh[k];
    g4[gi] = s;
  }
  const float ig = sigf(g4[0]);
  const float fg = sigf(g4[1]);
  const float gg = tanhf(g4[2]);
  const float og = sigf(g4[3]);
  const float cv = fg * cst[idx] + ig * gg;
  cst[idx] = cv;
  h_new[idx] = og * tanhf(cv);
}

// e[n] = <x[n], h[batch[n]]>; running segment max via ordered-uint atomicMax
__global__ __launch_bounds__(256) void attn_dot(
    const float* __restrict__ X, const float* __restrict__ h,
    const int* __restrict__ batch, float* __restrict__ evec,
    unsigned* __restrict__ emax) {
  const int lane = threadIdx.x & 31;
  const int n = blockIdx.x * 8 + (threadIdx.x >> 5);
  const int b = batch[n];
  const float4 xv = *(const float4*)(X + (size_t)n * HH + lane * 4);
  const float4 hv = *(const float4*)(h + (size_t)b * HH + lane * 4);
  float d = xv.x * hv.x + xv.y * hv.y + xv.z * hv.z + xv.w * hv.w;
#pragma unroll
  for (int off = 16; off > 0; off >>= 1) d += __shfl_xor(d, off, 32);
  if (lane == 0) {
    evec[n] = d;
    atomicMax(&emax[b], fenc(d));
  }
}

// w = exp(e - emax[b]); rnum[b] += w*x[n]; ssum[b] += w  (normalize later)
__global__ __launch_bounds__(256) void attn_acc(
    const float* __restrict__ X, const float* __restrict__ evec,
    const int* __restrict__ batch, const unsigned* __restrict__ emax,
    float* __restrict__ ssum, float* __restrict__ rnum) {
  const int lane = threadIdx.x & 31;
  const int n = blockIdx.x * 8 + (threadIdx.x >> 5);
  const int b = batch[n];
  const float w = expf(evec[n] - fdec(emax[b]));
  const float4 xv = *(const float4*)(X + (size_t)n * HH + lane * 4);
  float* dst = rnum + (size_t)b * HH + lane * 4;
  atomicAdd(dst + 0, w * xv.x);
  atomicAdd(dst + 1, w * xv.y);
  atomicAdd(dst + 2, w * xv.z);
  atomicAdd(dst + 3, w * xv.w);
  if (lane == 0) atomicAdd(&ssum[b], w);
}

// q_star = [h, rnum/(ssum+1e-16)]
__global__ __launch_bounds__(256) void qstar_update(
    const float* __restrict__ h_new, const float* __restrict__ rnum,
    const float* __restrict__ ssum, float* __restrict__ qstar) {
  const int idx = blockIdx.x * 256 + threadIdx.x;  // 0..8191
  const int b = idx >> 7;
  const int j = idx & (HH - 1);
  const float r = rnum[idx] / (ssum[b] + 1e-16f);
  qstar[(size_t)b * 2 * HH + j] = h_new[idx];
  qstar[(size_t)b * 2 * HH + HH + j] = r;
}

// out[b] = <h_final[b], Wout> + bout
__global__ __launch_bounds__(256) void readout(
    const float* __restrict__ h, const float* __restrict__ Wout,
    const float* __restrict__ bout, float* __restrict__ out) {
  const int lane = threadIdx.x & 31;
  const int b = blockIdx.x * 8 + (threadIdx.x >> 5);
  const float4 hv = *(const float4*)(h + (size_t)b * HH + lane * 4);
  const float4 wv = *(const float4*)(Wout + lane * 4);
  float d = hv.x * wv.x + hv.y * wv.y + hv.z * wv.z + hv.w * wv.w;
#pragma unroll
  for (int off = 16; off > 0; off >>= 1) d += __shfl_xor(d, off, 32);
  if (lane == 0) out[b] = d + bout[0];
}

// ---------------------------------------------------------------------------
extern "C" void kernel_launch(void* const* d_in, const int* in_sizes, int n_in,
                              void* d_out, int out_size, void* d_ws,
                              size_t ws_size, hipStream_t stream) {
  (void)in_sizes; (void)n_in; (void)out_size; (void)ws_size;
  const float* node_features = (const float*)d_in[0];
  const float* edge_features = (const float*)d_in[1];
  const int* Esrc = (const int*)d_in[2];
  const int* Etgt = (const int*)d_in[3];
  const int* batch = (const int*)d_in[4];
  const float* Win = (const float*)d_in[5];
  const float* bin = (const float*)d_in[6];
  const float* Wee = (const float*)d_in[7];
  const float* bee = (const float*)d_in[8];
  const float* Wg = (const float*)d_in[9];
  const float* bg = (const float*)d_in[10];
  const float* Wih = (const float*)d_in[11];
  const float* Whh = (const float*)d_in[12];
  const float* bih = (const float*)d_in[13];
  const float* bhh = (const float*)d_in[14];
  const float* Wout = (const float*)d_in[15];
  const float* bout = (const float*)d_in[16];

  char* p = (char*)d_ws;
  auto take = [&](size_t bytes) -> char* {
    char* r = p;
    p += (bytes + 255) & ~(size_t)255;
    return r;
  };
  float* xbuf = (float*)take((size_t)NN * HH * 4);   // node state (L2-resident)
  float* ybuf = (float*)take((size_t)NN * HH * 4);   // x @ Wg[l]
  float* accb = (float*)take((size_t)NN * HH * 4);   // segment-sum accumulator
  float* evec = (float*)take((size_t)NN * 4);
  float* h0 = (float*)take((size_t)BB * HH * 4);
  float* h1 = (float*)take((size_t)BB * HH * 4);
  float* cst = (float*)take((size_t)BB * HH * 4);
  float* qst = (float*)take((size_t)BB * 2 * HH * 4);
  float* rnum = (float*)take((size_t)BB * HH * 4);
  float* ssum = (float*)take((size_t)BB * 4);
  unsigned* emax = (unsigned*)take((size_t)BB * 4);

  hipMemsetAsync(h0, 0, (size_t)BB * HH * 4, stream);
  hipMemsetAsync(cst, 0, (size_t)BB * HH * 4, stream);
  hipMemsetAsync(qst, 0, (size_t)BB * 2 * HH * 4, stream);

  // x = node_features @ Win + bin   (f32 WMMA)
  gemm_xw_wmma<NODE_FD><<<dim3(NN / 128, HH / 16), 256, 0, stream>>>(
      node_features, Win, bin, xbuf);

  // 3 gated graph-conv layers
  for (int l = 0; l < LL; ++l) {
    gemm_xw_wmma<HH><<<dim3(NN / 128, HH / 16), 256, 0, stream>>>(
        xbuf, Wg + (size_t)l * HH * HH, nullptr, ybuf);
    hipMemsetAsync(accb, 0, (size_t)NN * HH * 4, stream);
    edge_msg<<<EE / 8, 256, 0, stream>>>(ybuf, edge_features, Esrc, Etgt, Wee,
                                         bee, accb);
    bias_act<<<2048, 256, 0, stream>>>(accb, bg + l * HH, xbuf,
                                       (l < LL - 1) ? 1 : 0);
  }

  // Set2Set readout, T sequential steps (ping-pong h)
  float* hOld = h0;
  float* hNew = h1;
  for (int t = 0; t < TT; ++t) {
    lstm_step<<<(BB * HH) / 256, 256, 0, stream>>>(
        qst, hOld, hNew, cst, Wih, Whh, bih, bhh, emax, ssum, rnum);
    attn_dot<<<NN / 8, 256, 0, stream>>>(xbuf, hNew, batch, evec, emax);
    attn_acc<<<NN / 8, 256, 0, stream>>>(xbuf, evec, batch, emax, ssum, rnum);
    qstar_update<<<(BB * HH) / 256, 256, 0, stream>>>(hNew, rnum, ssum, qst);
    float* tmp = hOld; hOld = hNew; hNew = tmp;
  }

  readout<<<BB / 8, 256, 0, stream>>>(hOld, Wout, bout, (float*)d_out);
}